// PhaseShift_3058016715100
// MI455X (gfx1250) — compile-verified
//
#include <hip/hip_runtime.h>
#include <cmath>
#include <cstdint>

// PhaseShift: phase = atan2(xi, xr); diff vs channel 0; np.unwrap along F axis.
// Shapes: [N=16, CH=4, F=513, T=512], f32. Memory-bound streaming scan along F.
// Thread = one (n, ch in 1..3, t) column; lanes = consecutive t -> coalesced
// 128B wave-rows. Inputs streamed via CDNA5 async global->LDS loads with an
// 8-deep circular pipeline (s_wait_asynccnt), falling back to unrolled direct
// loads + global_prefetch_b8 if the async builtin is unavailable.

#define AS1 __attribute__((address_space(1)))
#define AS3 __attribute__((address_space(3)))

#ifndef __has_builtin
#define __has_builtin(x) 0
#endif

#if defined(__HIP_DEVICE_COMPILE__) && __has_builtin(__builtin_amdgcn_global_load_async_to_lds_b32)
#define HAVE_ASYNC_LDS 1
#else
#define HAVE_ASYNC_LDS 0
#endif

constexpr int N_ = 16, CH_ = 4, F_ = 513, T_ = 512;
constexpr int FT_ = F_ * T_;  // 262656

__device__ __forceinline__ float unwrap_step(float p, float& prev, float& corr) {
  const float PIF    = 3.14159265358979323846f;
  const float TWOPI  = 6.28318530717958647692f;
  const float INV2PI = 0.15915494309189533577f;
  float dd = p - prev;
  float x  = dd + PIF;
  float m  = x - floorf(x * INV2PI) * TWOPI;   // jnp.mod(dd+pi, 2pi)
  float ddmod = m - PIF;
  if ((ddmod == -PIF) && (dd > 0.0f)) ddmod = PIF;
  float pc = (fabsf(dd) < PIF) ? 0.0f : (ddmod - dd);
  corr += pc;
  prev = p;
  return p + corr;
}

// Channel-0 output plane is unwrap(0) == 0 everywhere; write it explicitly.
__global__ void zero_ch0_kernel(float* __restrict__ out) {
  const int Q = FT_ / 4;  // 65664 float4 per plane
  int gid = blockIdx.x * blockDim.x + threadIdx.x;   // exactly N_*Q threads
  int n = gid / Q, r = gid % Q;
  float4 z = make_float4(0.f, 0.f, 0.f, 0.f);
  ((float4*)(out + (size_t)n * CH_ * FT_))[r] = z;
}

#if HAVE_ASYNC_LDS
#if __has_builtin(__builtin_amdgcn_s_wait_asynccnt)
#define WAIT_ASYNC(n) do { __builtin_amdgcn_s_wait_asynccnt(n); asm volatile("" ::: "memory"); } while (0)
#else
#define WAIT_ASYNC(n) asm volatile("s_wait_asynccnt %0" :: "i"(n) : "memory")
#endif
#endif

__global__ void __launch_bounds__(128)
unwrap_kernel(const float* __restrict__ xr, const float* __restrict__ xi,
              float* __restrict__ out) {
  int gid = blockIdx.x * blockDim.x + threadIdx.x;
  int t   = gid & (T_ - 1);      // lanes -> consecutive t (coalesced)
  int rem = gid >> 9;            // n*3 + chi   (wave-uniform: 512 % 32 == 0)
  int chi = rem % 3, n = rem / 3;
  size_t baseC = ((size_t)(n * CH_ + chi + 1)) * FT_ + t;
  size_t base0 = ((size_t)(n * CH_)) * FT_ + t;
  const float* rcp = xr + baseC;
  const float* icp = xi + baseC;
  const float* r0p = xr + base0;
  const float* i0p = xi + base0;
  float* o = out + baseC;

  // f = 0: out = phase diff itself, correction starts at 0.
  float p0 = atan2f(icp[0], rcp[0]) - atan2f(i0p[0], r0p[0]);
  o[0] = p0;
  float prev = p0, corr = 0.0f;

#if HAVE_ASYNC_LDS
  // 8-slot per-wave circular staging: 4 streams x 32 lanes x 4B per f.
  __shared__ float stage[4][8][4][32];
  int lane = threadIdx.x & 31;
  int w    = threadIdx.x >> 5;

  auto issue = [&](int f) {
    int sl = (f - 1) & 7;
    size_t off = (size_t)f * T_;
    __builtin_amdgcn_global_load_async_to_lds_b32(
        (AS1 int*)(uintptr_t)(rcp + off),
        (AS3 int*)(unsigned)(uintptr_t)&stage[w][sl][0][lane], 0, 0);
    __builtin_amdgcn_global_load_async_to_lds_b32(
        (AS1 int*)(uintptr_t)(icp + off),
        (AS3 int*)(unsigned)(uintptr_t)&stage[w][sl][1][lane], 0, 0);
    __builtin_amdgcn_global_load_async_to_lds_b32(
        (AS1 int*)(uintptr_t)(r0p + off),
        (AS3 int*)(unsigned)(uintptr_t)&stage[w][sl][2][lane], 0, 0);
    __builtin_amdgcn_global_load_async_to_lds_b32(
        (AS1 int*)(uintptr_t)(i0p + off),
        (AS3 int*)(unsigned)(uintptr_t)&stage[w][sl][3][lane], 0, 0);
  };

#pragma unroll
  for (int f = 1; f <= 8; ++f) issue(f);        // 32 async b32 in flight

  for (int f = 1; f <= 504; ++f) {
    WAIT_ASYNC(28);                             // oldest slot (f) landed
    int sl = (f - 1) & 7;
    float a = stage[w][sl][0][lane];
    float b = stage[w][sl][1][lane];
    float c = stage[w][sl][2][lane];
    float d = stage[w][sl][3][lane];
    asm volatile("s_wait_dscnt 0" ::: "memory"); // reads done before slot reuse
    issue(f + 8);                               // refills exactly slot sl
    float p = atan2f(b, a) - atan2f(d, c);
    __builtin_nontemporal_store(unwrap_step(p, prev, corr), o + (size_t)f * T_);
  }
  WAIT_ASYNC(0);                                // drain
  for (int f = 505; f <= 512; ++f) {
    int sl = (f - 1) & 7;
    float p = atan2f(stage[w][sl][1][lane], stage[w][sl][0][lane]) -
              atan2f(stage[w][sl][3][lane], stage[w][sl][2][lane]);
    __builtin_nontemporal_store(unwrap_step(p, prev, corr), o + (size_t)f * T_);
  }
#else
  // Fallback: unroll-by-4 direct loads + gfx1250 global_prefetch_b8 ahead.
  for (int f = 1; f <= 512; f += 4) {
    if (f + 32 <= 512) {
      __builtin_prefetch(rcp + (size_t)(f + 32) * T_, 0, 0);
      __builtin_prefetch(icp + (size_t)(f + 32) * T_, 0, 0);
      __builtin_prefetch(r0p + (size_t)(f + 32) * T_, 0, 0);
      __builtin_prefetch(i0p + (size_t)(f + 32) * T_, 0, 0);
    }
    float a[4], b[4], c[4], d[4];
#pragma unroll
    for (int j = 0; j < 4; ++j) {
      size_t off = (size_t)(f + j) * T_;
      a[j] = __builtin_nontemporal_load(rcp + off);
      b[j] = __builtin_nontemporal_load(icp + off);
      c[j] = __builtin_nontemporal_load(r0p + off);
      d[j] = __builtin_nontemporal_load(i0p + off);
    }
#pragma unroll
    for (int j = 0; j < 4; ++j) {
      float p = atan2f(b[j], a[j]) - atan2f(d[j], c[j]);
      __builtin_nontemporal_store(unwrap_step(p, prev, corr),
                                  o + (size_t)(f + j) * T_);
    }
  }
#endif
}

extern "C" void kernel_launch(void* const* d_in, const int* in_sizes, int n_in,
                              void* d_out, int out_size, void* d_ws, size_t ws_size,
                              hipStream_t stream) {
  (void)in_sizes; (void)n_in; (void)out_size; (void)d_ws; (void)ws_size;
  const float* xr = (const float*)d_in[0];
  const float* xi = (const float*)d_in[1];
  float* out = (float*)d_out;

  // 16 * (262656/4) = 1,050,624 float4 stores -> 4104 blocks of 256, exact.
  zero_ch0_kernel<<<(N_ * (FT_ / 4)) / 256, 256, 0, stream>>>(out);

  // 16 * 3 * 512 = 24,576 threads -> 192 blocks of 128 (768 waves), exact.
  unwrap_kernel<<<(N_ * 3 * T_) / 128, 128, 0, stream>>>(xr, xi, out);
}